// GraphVAE_88476326298126
// MI455X (gfx1250) — compile-verified
//
#include <hip/hip_runtime.h>
#include <math.h>

// ---------------------------------------------------------------------------
// GraphVAE (SplineConv GNN) for MI455X / gfx1250.
// Memory-bound (~2.8 GB traffic vs ~20 GFLOP) -> full fp32 precision with
// V_WMMA_F32_16X16X4_F32 for all node-side GEMMs. GEMM inner loop is
// guard-free: K padded to x4, M padded to x64, B packed in a quad layout so
// every fragment is a single global_load_b64.
// ---------------------------------------------------------------------------

#define CIN   3
#define HID   64
#define LAT   32
#define KK    3
#define K3    27
#define SLOPE 0.01f
#define MT    4      // 4 x 16 output columns per wave (A-fragment reuse)

typedef __attribute__((ext_vector_type(2))) float v2f;
typedef __attribute__((ext_vector_type(8))) float v8f;

// ---------------- float <-> order-preserving uint --------------------------
__device__ __forceinline__ unsigned enc_ord(float f) {
    unsigned u = __float_as_uint(f);
    return (u & 0x80000000u) ? ~u : (u | 0x80000000u);
}
__device__ __forceinline__ float dec_ord(unsigned u) {
    u = (u & 0x80000000u) ? (u & 0x7FFFFFFFu) : ~u;
    return __uint_as_float(u);
}

// ---------------- init: mm[0..2]=min enc, mm[3..5]=max enc -----------------
__global__ void init_minmax_kernel(unsigned* mm) {
    int t = threadIdx.x;
    if (t < 3) mm[t] = 0xFFFFFFFFu;
    else if (t < 6) mm[t] = 0u;
}

// ---------------- per-dim min/max of (x[src]-x[dst]) over edges ------------
__global__ void minmax_kernel(const float* __restrict__ x,
                              const int* __restrict__ src,
                              const int* __restrict__ dst,
                              unsigned* __restrict__ mm, int E) {
    __shared__ unsigned smin[3], smax[3];
    if (threadIdx.x < 3) { smin[threadIdx.x] = 0xFFFFFFFFu; smax[threadIdx.x] = 0u; }
    __syncthreads();
    int e = blockIdx.x * blockDim.x + threadIdx.x;
    if (e < E) {
        int s = src[e], d = dst[e];
        #pragma unroll
        for (int dd = 0; dd < 3; ++dd) {
            float diff = x[s * CIN + dd] - x[d * CIN + dd];
            unsigned u = enc_ord(diff);
            atomicMin(&smin[dd], u);   // ds_min_u32
            atomicMax(&smax[dd], u);   // ds_max_u32
        }
    }
    __syncthreads();
    if (threadIdx.x < 3) {
        atomicMin(&mm[threadIdx.x],     smin[threadIdx.x]);
        atomicMax(&mm[3 + threadIdx.x], smax[threadIdx.x]);
    }
}

// ---------------- spline basis + taps per edge, plus degree ----------------
__global__ void edge_basis_kernel(const float* __restrict__ x,
                                  const int* __restrict__ src,
                                  const int* __restrict__ dst,
                                  const unsigned* __restrict__ mm,
                                  float* __restrict__ basis,
                                  int* __restrict__ eidx,
                                  float* __restrict__ deg, int E) {
    int e = blockIdx.x * blockDim.x + threadIdx.x;
    if (e >= E) return;
    int s = src[e], d = dst[e];
    float b[8] = {1.f,1.f,1.f,1.f,1.f,1.f,1.f,1.f};
    int   id[8] = {0,0,0,0,0,0,0,0};
    int stride = 1;
    #pragma unroll
    for (int dd = 0; dd < 3; ++dd) {
        float diff = x[s * CIN + dd] - x[d * CIN + dd];
        float pmin = dec_ord(mm[dd]);
        float pmax = dec_ord(mm[3 + dd]);
        float p    = (diff - pmin) / (pmax - pmin + 1e-8f);
        float pos  = p * (float)(KK - 1);
        float lof  = floorf(pos);
        float frac = pos - lof;
        int lo = (int)lof; lo = lo < 0 ? 0 : (lo > KK - 1 ? KK - 1 : lo);
        int hi = (lo + 1 > KK - 1) ? KK - 1 : lo + 1;
        #pragma unroll
        for (int ss = 0; ss < 8; ++ss) {
            int bit = (ss >> dd) & 1;
            b[ss]  *= bit ? frac : (1.0f - frac);
            id[ss] += (bit ? hi : lo) * stride;
        }
        stride *= KK;
    }
    #pragma unroll
    for (int ss = 0; ss < 8; ++ss) {
        basis[e * 8 + ss] = b[ss];
        eidx [e * 8 + ss] = id[ss];
    }
    atomicAdd(&deg[d], 1.0f);
}

// ---------------- zero-pad x (N x 3) -> xpad (N x 4) -----------------------
__global__ void pad_x_kernel(const float* __restrict__ x, float* __restrict__ xp, int N) {
    int t = blockIdx.x * blockDim.x + threadIdx.x;
    if (t >= N * 4) return;
    int n = t >> 2, d = t & 3;
    xp[t] = (d < CIN) ? x[n * CIN + d] : 0.f;
}

// ---------------- pack spline weights into quad-fragment layout ------------
// Bq[q*(Mp*4) + c*4 + kk] = Blogical[i = q*4+kk][c], zero-padded.
// Blogical[i][c] = (c/co < K3) ? w[( (c/co)*ci + i )*co + c%co] : root[i*co + c%co]
__global__ void pack_w_quad_kernel(const float* __restrict__ w,
                                   const float* __restrict__ root,
                                   float* __restrict__ Bq,
                                   int ci, int co, int CT, int Mp, int Kp) {
    int t = blockIdx.x * blockDim.x + threadIdx.x;
    if (t >= Kp * Mp) return;
    int q  = t / (Mp * 4);
    int rm = t % (Mp * 4);
    int c  = rm >> 2;
    int kk = rm & 3;
    int i  = q * 4 + kk;
    float v = 0.f;
    if (i < ci && c < CT) {
        int k = c / co, o = c % co;
        v = (k < K3) ? w[(k * ci + i) * co + o] : root[i * co + o];
    }
    Bq[t] = v;
}

// ---------------- pack generic row-major B (K x M) into quad layout --------
__global__ void pack_b_quad_kernel(const float* __restrict__ B,
                                   float* __restrict__ Bq,
                                   int K, int M, int Mp, int Kp) {
    int t = blockIdx.x * blockDim.x + threadIdx.x;
    if (t >= Kp * Mp) return;
    int q  = t / (Mp * 4);
    int rm = t % (Mp * 4);
    int c  = rm >> 2;
    int kk = rm & 3;
    int i  = q * 4 + kk;
    Bq[t] = (i < K && c < M) ? B[i * M + c] : 0.f;
}

// ---------------- fp32 WMMA GEMM: C(NxMp) = A(NxK) @ Bq, guard-free --------
// Requirements: K % 4 == 0, Mp % (16*MT) == 0. One wave per block computes a
// 16 x 64 output strip. A fragment (ISA 7.12.2, 32-bit A 16x4): lanes 0-15
// hold K=k0,k0+1 of row M=lane; lanes 16-31 hold K=k0+2,k0+3. C/D layout:
// VGPR r -> M = r + 8*(lane>=16), N = lane&15.
__global__ void gemm_wmma_kernel(const float* __restrict__ A,
                                 const float* __restrict__ Bq,
                                 float* __restrict__ C,
                                 int N, int K, int Mp) {
    int tm   = blockIdx.x * 16;
    int tn0  = blockIdx.y * 16 * MT;
    int lane = threadIdx.x;
    int half = lane >> 4;
    int l16  = lane & 15;

    v8f acc[MT];
    #pragma unroll
    for (int j = 0; j < MT; ++j) acc[j] = (v8f){0.f,0.f,0.f,0.f,0.f,0.f,0.f,0.f};

    int arow = tm + l16;
    arow = arow < N ? arow : N - 1;                 // branchless clamp
    const float* Arow = A + (size_t)arow * K + 2 * half;
    const float* Bcol = Bq + (size_t)(tn0 + l16) * 4 + 2 * half;
    const size_t bstep = (size_t)Mp * 4;

    for (int k0 = 0; k0 < K; k0 += 4) {
        v2f a = *(const v2f*)(Arow + k0);           // global_load_b64
        #pragma unroll
        for (int j = 0; j < MT; ++j) {
            v2f b = *(const v2f*)(Bcol + (size_t)(k0 >> 2) * bstep + j * 64);
            acc[j] = __builtin_amdgcn_wmma_f32_16x16x4_f32(
                false, a, false, b, (short)0, acc[j], false, false);
        }
    }

    int crow0 = tm + 8 * half;
    #pragma unroll
    for (int j = 0; j < MT; ++j) {
        int ccol = tn0 + j * 16 + l16;
        #pragma unroll
        for (int r = 0; r < 8; ++r) {
            int crow = crow0 + r;
            if (crow < N) C[(size_t)crow * Mp + ccol] = acc[j][r];
        }
    }
}

// ---------------- edge gather + weighted sum + scatter ---------------------
__global__ void edge_msg_kernel(const float* __restrict__ y,
                                const float* __restrict__ basis,
                                const int* __restrict__ eidx,
                                const int* __restrict__ src,
                                const int* __restrict__ dst,
                                float* __restrict__ agg,
                                int E, int co, int CTp) {
    int t = blockIdx.x * blockDim.x + threadIdx.x;
    if (t >= E * co) return;
    int e = t / co, o = t % co;
    const float* yb = y + (size_t)src[e] * CTp;
    float m = 0.f;
    #pragma unroll
    for (int s = 0; s < 8; ++s)
        m += basis[e * 8 + s] * yb[eidx[e * 8 + s] * co + o];
    atomicAdd(&agg[dst[e] * co + o], m);
}

// ---------------- agg/deg + root-term + bias (+ leaky) ---------------------
__global__ void finalize_kernel(const float* __restrict__ agg,
                                const float* __restrict__ deg,
                                const float* __restrict__ ybuf,
                                const float* __restrict__ bias,
                                float* __restrict__ out,
                                int N, int co, int CTp, int doLeaky) {
    int t = blockIdx.x * blockDim.x + threadIdx.x;
    if (t >= N * co) return;
    int n = t / co, o = t % co;
    float dg = deg[n]; dg = dg < 1.0f ? 1.0f : dg;
    float v = agg[t] / dg + ybuf[(size_t)n * CTp + K3 * co + o] + bias[o];
    if (doLeaky) v = v > 0.f ? v : SLOPE * v;
    out[t] = v;
}

// ---------------- VAE reparameterization -----------------------------------
__global__ void vae_split_kernel(const float* __restrict__ h3,
                                 const float* __restrict__ eps,
                                 float* __restrict__ mu_out,
                                 float* __restrict__ logvar_out,
                                 float* __restrict__ z, int N) {
    int t = blockIdx.x * blockDim.x + threadIdx.x;
    if (t >= N * LAT) return;
    int n = t / LAT, l = t % LAT;
    float mu = h3[n * (2 * LAT) + l];
    float lv = h3[n * (2 * LAT) + LAT + l];
    mu_out[t]     = mu;
    logvar_out[t] = lv;
    z[t] = mu + eps[t] * expf(0.5f * lv);
}

__global__ void bias_leaky_kernel(float* __restrict__ h,
                                  const float* __restrict__ b,
                                  int N, int co) {
    int t = blockIdx.x * blockDim.x + threadIdx.x;
    if (t >= N * co) return;
    float v = h[t] + b[t % co];
    h[t] = v > 0.f ? v : SLOPE * v;
}

// ---------------------------------------------------------------------------
static inline size_t alignUp(size_t v) { return (v + 255) & ~(size_t)255; }
static inline int roundUp(int v, int m) { return (v + m - 1) / m * m; }

extern "C" void kernel_launch(void* const* d_in, const int* in_sizes, int n_in,
                              void* d_out, int out_size, void* d_ws, size_t ws_size,
                              hipStream_t stream) {
    const int Nn = in_sizes[0] / CIN;       // 30000
    const int Ee = in_sizes[1] / 2;         // 240000

    const float* x     = (const float*)d_in[0];
    const int*   src   = (const int*)d_in[1];
    const int*   dst   = src + Ee;
    const float* w1 = (const float*)d_in[2],  *r1 = (const float*)d_in[3],  *b1 = (const float*)d_in[4];
    const float* w2 = (const float*)d_in[5],  *r2 = (const float*)d_in[6],  *b2 = (const float*)d_in[7];
    const float* w3 = (const float*)d_in[8],  *r3 = (const float*)d_in[9],  *b3 = (const float*)d_in[10];
    const float* fc_w = (const float*)d_in[11], *fc_b = (const float*)d_in[12];
    const float* w4 = (const float*)d_in[13], *r4 = (const float*)d_in[14], *b4 = (const float*)d_in[15];
    const float* w5 = (const float*)d_in[16], *r5 = (const float*)d_in[17], *b5 = (const float*)d_in[18];
    const float* eps = (const float*)d_in[19];

    float* recon      = (float*)d_out;                    // N*CIN
    float* mu_out     = recon + (size_t)Nn * CIN;         // N*LAT
    float* logvar_out = mu_out + (size_t)Nn * LAT;        // N*LAT

    // ---- workspace carve-up ----
    const int CT_MAX = K3 * HID + HID;   // 1792 (already % 64 == 0)
    char* p = (char*)d_ws;
    auto take = [&](size_t bytes) { char* q = p; p += alignUp(bytes); return q; };
    float*    ybuf  = (float*)   take((size_t)Nn * CT_MAX * sizeof(float));   // 215 MB
    float*    wcat  = (float*)   take((size_t)HID * CT_MAX * sizeof(float));
    float*    basis = (float*)   take((size_t)Ee * 8 * sizeof(float));
    int*      eidx  = (int*)     take((size_t)Ee * 8 * sizeof(int));
    float*    agg   = (float*)   take((size_t)Nn * HID * sizeof(float));
    float*    deg   = (float*)   take((size_t)Nn * sizeof(float));
    unsigned* mm    = (unsigned*)take(6 * sizeof(unsigned));
    float*    xpad  = (float*)   take((size_t)Nn * 4 * sizeof(float));
    float*    h1    = (float*)   take((size_t)Nn * HID * sizeof(float));
    float*    h2    = (float*)   take((size_t)Nn * HID * sizeof(float));
    float*    h3    = (float*)   take((size_t)Nn * HID * sizeof(float));
    float*    zbuf  = (float*)   take((size_t)Nn * LAT * sizeof(float));
    float*    hfc   = (float*)   take((size_t)Nn * HID * sizeof(float));
    float*    h4    = (float*)   take((size_t)Nn * HID * sizeof(float));
    (void)ws_size; (void)n_in; (void)out_size;

    // ---- preprocessing: min/max, basis, degree, padded x ----
    hipLaunchKernelGGL(init_minmax_kernel, dim3(1), dim3(32), 0, stream, mm);
    hipMemsetAsync(deg, 0, (size_t)Nn * sizeof(float), stream);
    hipLaunchKernelGGL(minmax_kernel, dim3((Ee + 255) / 256), dim3(256), 0, stream,
                       x, src, dst, mm, Ee);
    hipLaunchKernelGGL(edge_basis_kernel, dim3((Ee + 255) / 256), dim3(256), 0, stream,
                       x, src, dst, mm, basis, eidx, deg, Ee);
    hipLaunchKernelGGL(pad_x_kernel, dim3((Nn * 4 + 255) / 256), dim3(256), 0, stream,
                       x, xpad, Nn);

    // ---- one spline-conv layer (A has Kpad columns, Kpad % 4 == 0) --------
    auto run_layer = [&](const float* xin, int ci, int Kpad, int co,
                         const float* w, const float* root, const float* bias,
                         float* out, bool leaky) {
        int CT  = K3 * co + co;
        int CTp = roundUp(CT, 16 * MT);
        hipLaunchKernelGGL(pack_w_quad_kernel,
                           dim3((Kpad * CTp + 255) / 256), dim3(256), 0, stream,
                           w, root, wcat, ci, co, CT, CTp, Kpad);
        dim3 gg((Nn + 15) / 16, CTp / (16 * MT));
        hipLaunchKernelGGL(gemm_wmma_kernel, gg, dim3(32), 0, stream,
                           xin, wcat, ybuf, Nn, Kpad, CTp);
        hipMemsetAsync(agg, 0, (size_t)Nn * co * sizeof(float), stream);
        hipLaunchKernelGGL(edge_msg_kernel,
                           dim3(((size_t)Ee * co + 255) / 256), dim3(256), 0, stream,
                           ybuf, basis, eidx, src, dst, agg, Ee, co, CTp);
        hipLaunchKernelGGL(finalize_kernel, dim3((Nn * co + 255) / 256), dim3(256), 0, stream,
                           agg, deg, ybuf, bias, out, Nn, co, CTp, leaky ? 1 : 0);
    };

    run_layer(xpad, CIN, 4,  HID,     w1, r1, b1, h1, true);   // layer 1 + leaky
    run_layer(h1,   HID, 64, HID,     w2, r2, b2, h2, true);   // layer 2 + leaky
    run_layer(h2,   HID, 64, 2 * LAT, w3, r3, b3, h3, false);  // layer 3 (mu|logvar)

    // ---- reparameterize + fc ----
    hipLaunchKernelGGL(vae_split_kernel, dim3((Nn * LAT + 255) / 256), dim3(256), 0, stream,
                       h3, eps, mu_out, logvar_out, zbuf, Nn);
    {   // hfc = z (N x LAT) @ fc_w (LAT x HID); M=64 -> one tile column
        hipLaunchKernelGGL(pack_b_quad_kernel,
                           dim3((LAT * HID + 255) / 256), dim3(256), 0, stream,
                           fc_w, wcat, LAT, HID, HID, LAT);
        dim3 gg((Nn + 15) / 16, HID / (16 * MT));
        hipLaunchKernelGGL(gemm_wmma_kernel, gg, dim3(32), 0, stream,
                           zbuf, wcat, hfc, Nn, LAT, HID);
    }
    hipLaunchKernelGGL(bias_leaky_kernel, dim3((Nn * HID + 255) / 256), dim3(256), 0, stream,
                       hfc, fc_b, Nn, HID);

    run_layer(hfc, HID, 64, HID, w4, r4, b4, h4, true);        // layer 4 + leaky
    run_layer(h4,  HID, 64, CIN, w5, r5, b5, recon, false);    // layer 5 -> recon
}